// AttentionModule_2611340115974
// MI455X (gfx1250) — compile-verified
//
#include <hip/hip_runtime.h>

#define BSZ 64
#define SEQ 1024
#define HID 1024
#define OUT 1024

#define NCHUNK 8            // seq chunks per batch
#define CHROWS 128          // rows per chunk
#define TROWS  16           // rows per LDS tile
#define NTILES (CHROWS / TROWS)

typedef __attribute__((ext_vector_type(2))) float v2f;
typedef __attribute__((ext_vector_type(4))) float v4f;
typedef __attribute__((ext_vector_type(8))) float v8f;

// ---------------------------------------------------------------------------
// CDNA5 async global->LDS copy (ASYNCcnt-tracked). INST_OFFSET applies to both
// the LDS and the global address, so one VGPR pair + one LDS VGPR covers the
// whole per-thread slice.
// ---------------------------------------------------------------------------
template <int OFF>
__device__ __forceinline__ void async_copy_b128(unsigned lds_off,
                                                unsigned long long gaddr) {
  asm volatile("global_load_async_to_lds_b128 %0, %1, off offset:%2"
               :: "v"(lds_off), "v"(gaddr), "i"(OFF) : "memory");
}
__device__ __forceinline__ void wait_async_le(int) {}
__device__ __forceinline__ void wait_async_4() {
  asm volatile("s_wait_asynccnt 0x4" ::: "memory");
}
__device__ __forceinline__ void wait_async_0() {
  asm volatile("s_wait_asynccnt 0x0" ::: "memory");
}

__device__ __forceinline__ float waveSum(float v) {
  #pragma unroll
  for (int off = 16; off; off >>= 1) v += __shfl_xor(v, off, 32);
  return v;
}
__device__ __forceinline__ float waveMax(float v) {
  #pragma unroll
  for (int off = 16; off; off >>= 1) v = fmaxf(v, __shfl_xor(v, off, 32));
  return v;
}

// ---------------------------------------------------------------------------
// Kernel A: x = hidden @ W1^T  (M=64, N=1024, K=1024), f32 WMMA 16x16x4.
// One wave per 16x16 tile; layouts per CDNA5 ISA 7.12.2.
// ---------------------------------------------------------------------------
__global__ __launch_bounds__(32) void gemm_xw1(const float* __restrict__ A,
                                               const float* __restrict__ W,
                                               float* __restrict__ X) {
  const int mBase = (blockIdx.x & 3) * 16;
  const int nBase = (blockIdx.x >> 2) * 16;
  const int lane  = threadIdx.x & 31;
  const int rc    = lane & 15;
  const int koff  = (lane >> 4) << 1;

  v8f c = {};
  const float* ap = A + (mBase + rc) * HID + koff;
  const float* bp = W + (nBase + rc) * HID + koff;

  for (int k = 0; k < HID; k += 4) {
    v2f a = *(const v2f*)(ap + k);
    v2f b = *(const v2f*)(bp + k);
    c = __builtin_amdgcn_wmma_f32_16x16x4_f32(false, a, false, b, (short)0, c,
                                              false, false);
  }

  const int orow = (lane >> 4) * 8;
  #pragma unroll
  for (int r = 0; r < 8; ++r)
    X[(mBase + orow + r) * OUT + nBase + rc] = c[r];
}

// ---------------------------------------------------------------------------
// Kernel B1: per (chunk, batch) flash-style partial attention.
//   - stages enc rows through LDS with double-buffered async b128 copies
//     (enc is read from HBM exactly ONCE for both score and context passes)
//   - online softmax inside the chunk (local m, l, unnormalized ctx partial)
// Outputs to workspace: masked scores v[b][s], (m,l) per chunk, ctx partials.
// ---------------------------------------------------------------------------
__global__ __launch_bounds__(1024) void attn_part(
    const float* __restrict__ enc, const float* __restrict__ x,
    const int* __restrict__ src_lens,
    float* __restrict__ vws,    // [BSZ][SEQ] masked scores
    float* __restrict__ mlws,   // [BSZ][NCHUNK][2]  (m, l)
    float* __restrict__ ctxp) { // [BSZ][NCHUNK][OUT] unnormalized partials
  __shared__ __align__(16) float s_buf[2][TROWS * OUT]; // 2 x 64KB
  __shared__ float s_x[HID];
  __shared__ float s_pdot[32];
  __shared__ float s_p[TROWS];
  __shared__ float s_newM, s_psum;

  const int chunk = blockIdx.x;
  const int b     = blockIdx.y;
  const int tid   = threadIdx.x;
  const int wave  = tid >> 5;
  const int lane  = tid & 31;
  const int sChunk = chunk * CHROWS;
  const float* encb = enc + (size_t)b * SEQ * OUT;

  const unsigned lds_base0 = (unsigned)(size_t)(&s_buf[0][0]) + tid * 16u;
  const unsigned lds_base1 = (unsigned)(size_t)(&s_buf[1][0]) + tid * 16u;

  // issue tile 0 (16 rows = contiguous 64KB; per-thread slice = 4 x b128)
  {
    unsigned long long g =
        (unsigned long long)(size_t)(encb + (size_t)sChunk * OUT + tid * 4);
    async_copy_b128<0>(lds_base0, g);
    async_copy_b128<16384>(lds_base0, g);
    async_copy_b128<32768>(lds_base0, g);
    async_copy_b128<49152>(lds_base0, g);
  }

  s_x[tid] = x[b * HID + tid];
  const int len = src_lens[b];

  float Mloc = -3.0e38f, L = 0.f, acc = 0.f;

  for (int t = 0; t < NTILES; ++t) {
    // issue next tile into the other buffer, then wait for current tile
    if (t + 1 < NTILES) {
      const unsigned nb = ((t + 1) & 1) ? lds_base1 : lds_base0;
      unsigned long long g = (unsigned long long)(size_t)(
          encb + (size_t)(sChunk + (t + 1) * TROWS) * OUT + tid * 4);
      async_copy_b128<0>(nb, g);
      async_copy_b128<16384>(nb, g);
      async_copy_b128<32768>(nb, g);
      async_copy_b128<49152>(nb, g);
      wait_async_4();   // first 4 (tile t) complete; loads finish in order
    } else {
      wait_async_0();
    }
    __syncthreads();    // tile t visible to all waves

    const float* tb = &s_buf[t & 1][0];

    // ---- score dot: 2 waves per row (row = wave&15, half = wave>>4) ----
    {
      const int row  = wave & 15;
      const int half = wave >> 4;
      const float* rp = tb + row * OUT + half * 512;
      float d = 0.f;
      #pragma unroll
      for (int i = 0; i < 4; ++i) {
        const int o = i * 128 + lane * 4;
        v4f e  = *(const v4f*)(rp + o);
        v4f xv = *(const v4f*)(s_x + half * 512 + o);
        d += e.x * xv.x + e.y * xv.y + e.z * xv.z + e.w * xv.w;
      }
      d = waveSum(d);
      if (lane == 0) s_pdot[wave] = d;
    }
    __syncthreads();

    // ---- wave 0: mask, tile max, p = exp(v - newM), partial sum ----
    if (wave == 0) {
      float v = -3.0e38f;
      if (lane < TROWS) {
        const int s = sChunk + t * TROWS + lane;
        float sc = s_pdot[lane] + s_pdot[lane + 16];
        float mv = (s < len) ? sc : 0.0f;
        if (mv == 0.0f) mv = -1e10f;     // exact-zero -> -1e10 (ref semantics)
        vws[b * SEQ + s] = mv;
        v = mv;
      }
      const float newM = fmaxf(Mloc, waveMax(v));
      const float p = (lane < TROWS) ? __expf(v - newM) : 0.f;
      if (lane < TROWS) s_p[lane] = p;
      const float ps = waveSum(p);
      if (lane == 0) { s_newM = newM; s_psum = ps; }
    }
    __syncthreads();

    // ---- all threads: rescale running state, accumulate weighted columns ----
    {
      const float newM  = s_newM;
      const float scale = __expf(Mloc - newM);
      L   = L * scale + s_psum;
      acc = acc * scale;
      #pragma unroll
      for (int r = 0; r < TROWS; ++r)
        acc += s_p[r] * tb[r * OUT + tid];
      Mloc = newM;
    }
    __syncthreads();   // safe to overwrite this buffer next iteration
  }

  ctxp[((size_t)b * NCHUNK + chunk) * OUT + tid] = acc;
  if (tid == 0) {
    mlws[(b * NCHUNK + chunk) * 2 + 0] = Mloc;
    mlws[(b * NCHUNK + chunk) * 2 + 1] = L;
  }
}

// ---------------------------------------------------------------------------
// Kernel B2: combine chunk partials -> ctx and attn_scores.
// ---------------------------------------------------------------------------
__global__ __launch_bounds__(1024) void attn_combine(
    const float* __restrict__ vws, const float* __restrict__ mlws,
    const float* __restrict__ ctxp,
    float* __restrict__ attn_out,  // [SEQ][BSZ]
    float* __restrict__ ctx) {     // [BSZ][OUT]
  __shared__ float sm[NCHUNK], sl[NCHUNK];
  const int b = blockIdx.x;
  const int tid = threadIdx.x;

  if (tid < NCHUNK) {
    sm[tid] = mlws[(b * NCHUNK + tid) * 2 + 0];
    sl[tid] = mlws[(b * NCHUNK + tid) * 2 + 1];
  }
  __syncthreads();

  float M = -3.0e38f;
  #pragma unroll
  for (int i = 0; i < NCHUNK; ++i) M = fmaxf(M, sm[i]);
  float L = 0.f;
  #pragma unroll
  for (int i = 0; i < NCHUNK; ++i) L += sl[i] * __expf(sm[i] - M);
  const float invL = 1.0f / L;

  float c = 0.f;
  #pragma unroll
  for (int i = 0; i < NCHUNK; ++i)
    c += __expf(sm[i] - M) * ctxp[((size_t)b * NCHUNK + i) * OUT + tid];
  ctx[b * OUT + tid] = c * invL;

  attn_out[tid * BSZ + b] = __expf(vws[b * SEQ + tid] - M) * invL;
}

// ---------------------------------------------------------------------------
// Kernel C: out = tanh(concat(ctx, hidden) @ W2^T)  (M=64, N=1024, K=2048)
// ---------------------------------------------------------------------------
__global__ __launch_bounds__(32) void gemm_out(const float* __restrict__ Ctx,
                                               const float* __restrict__ Hid,
                                               const float* __restrict__ W2,
                                               float* __restrict__ Out) {
  const int mBase = (blockIdx.x & 3) * 16;
  const int nBase = (blockIdx.x >> 2) * 16;
  const int lane  = threadIdx.x & 31;
  const int rc    = lane & 15;
  const int koff  = (lane >> 4) << 1;
  const int K2    = HID + OUT;

  v8f c = {};
  const float* actx = Ctx + (mBase + rc) * HID + koff;
  const float* ahid = Hid + (mBase + rc) * HID + koff;
  const float* bp   = W2 + (nBase + rc) * K2 + koff;

  for (int k = 0; k < HID; k += 4) {
    v2f a = *(const v2f*)(actx + k);
    v2f b = *(const v2f*)(bp + k);
    c = __builtin_amdgcn_wmma_f32_16x16x4_f32(false, a, false, b, (short)0, c,
                                              false, false);
  }
  for (int k = 0; k < OUT; k += 4) {
    v2f a = *(const v2f*)(ahid + k);
    v2f b = *(const v2f*)(bp + HID + k);
    c = __builtin_amdgcn_wmma_f32_16x16x4_f32(false, a, false, b, (short)0, c,
                                              false, false);
  }

  const int orow = (lane >> 4) * 8;
  #pragma unroll
  for (int r = 0; r < 8; ++r)
    Out[(mBase + orow + r) * OUT + nBase + rc] = tanhf(c[r]);
}

// ---------------------------------------------------------------------------
extern "C" void kernel_launch(void* const* d_in, const int* in_sizes, int n_in,
                              void* d_out, int out_size, void* d_ws, size_t ws_size,
                              hipStream_t stream) {
  const float* hidden   = (const float*)d_in[0];  // [64,1024]
  const float* enc      = (const float*)d_in[1];  // [64,1024,1024]
  const int*   src_lens = (const int*)d_in[2];    // [64]
  const float* W1       = (const float*)d_in[3];  // [1024,1024]
  const float* W2       = (const float*)d_in[4];  // [1024,2048]

  float* out  = (float*)d_out;                 // [64,1024]
  float* attn = (float*)d_out + BSZ * OUT;     // [1024,64]

  float* x    = (float*)d_ws;                           // 65536
  float* ctx  = x + BSZ * OUT;                          // 65536
  float* vws  = ctx + BSZ * OUT;                        // 65536
  float* mlws = vws + BSZ * SEQ;                        // 1024
  float* ctxp = mlws + BSZ * NCHUNK * 2;                // 524288

  const int tiles = (BSZ / 16) * (OUT / 16);   // 256 waves

  gemm_xw1<<<tiles, 32, 0, stream>>>(hidden, W1, x);
  attn_part<<<dim3(NCHUNK, BSZ), 1024, 0, stream>>>(enc, x, src_lens,
                                                    vws, mlws, ctxp);
  attn_combine<<<BSZ, 1024, 0, stream>>>(vws, mlws, ctxp, attn, ctx);
  gemm_out<<<tiles, 32, 0, stream>>>(ctx, hidden, W2, out);
}